// polNormal_NoLayer_37005438222424
// MI455X (gfx1250) — compile-verified
//
#include <hip/hip_runtime.h>

typedef float v2f __attribute__((ext_vector_type(2)));
typedef float v8f __attribute__((ext_vector_type(8)));

#define BB 2
#define NN 131072
#define KK 9
#define PP 4
#define DD 4
// S=1, CI=8, CO=8, PDS=16, 128 output cols per point

__global__ __launch_bounds__(256) void polgrid_kernel(
    const float* __restrict__ x,      // (B,N,8)
    const float* __restrict__ dx,     // (B,N,9)
    const float* __restrict__ dy,     // (B,N,9)
    const int*   __restrict__ adj,    // (N,9)
    const float* __restrict__ phis,   // (4)
    const float* __restrict__ dists,  // (4)
    const float* __restrict__ sigma,  // (1)
    const float* __restrict__ amp,    // (16,8,8)
    float* __restrict__ out)          // (B,N,16,8)
{
    __shared__ float lds[8][16][128];   // 64 KB: one 16x512B out tile per wave

    const int lane  = threadIdx.x & 31;
    const int wave  = threadIdx.x >> 5;
    const int tile  = blockIdx.x * 8 + wave;   // 16 points per tile
    const int row   = lane & 15;
    const int khalf = lane >> 4;

    const long bn0 = (long)tile * 16;          // flattened b*N + n base
    const long bn  = bn0 + row;
    const long b   = bn / NN;
    const long n   = bn % NN;

    // ---- scalars ----
    float sig = fmaxf(sigma[0], 1e-10f);
    const float inv2s2 = -0.5f / (sig * sig);
    const float norm   = rsqrtf(2.0f * 3.14159265358979f * sig * sig);

    // ---- Gaussian centers ----
    float mul_[16], mut_[16];
    #pragma unroll
    for (int p = 0; p < PP; ++p) {
        const float ph = phis[p];
        const float cp = __cosf(ph);
        const float sp = __sinf(ph);
        #pragma unroll
        for (int d = 0; d < DD; ++d) {
            mul_[p*DD+d] = cp * dists[d];
            mut_[p*DD+d] = sp * dists[d];
        }
    }

    // ---- preload block-diagonal Amp into WMMA-B layout ----
    // tile j pairs pds {2j,2j+1}; 4 wmma steps (m) of K=4; lane holds k = m*4 + 2*khalf + {0,1}
    float ampB[64];
    {
        const int ncol = lane & 15;
        const int pcol = ncol >> 3;          // which pds of the pair this column is
        const int o    = ncol & 7;
        #pragma unroll
        for (int j = 0; j < 8; ++j)
          #pragma unroll
          for (int m = 0; m < 4; ++m)
            #pragma unroll
            for (int idx = 0; idx < 2; ++idx) {
                const int k4 = m*4 + khalf*2 + idx;   // local K 0..15
                const int pk = k4 >> 3;               // pds of pair supplying this K
                const int c  = k4 & 7;
                ampB[(j*4 + m)*2 + idx] =
                    (pk == pcol) ? amp[((2*j + pcol)*8 + c)*8 + o] : 0.0f;
            }
    }

    // ---- stage 1: y[row,(pds,c)] accumulated straight into WMMA-A layout ----
    // ya[chunk*2+idx] holds K index kk = chunk*4 + 2*khalf + idx, kk = pds*8 + c
    float ya[64];
    #pragma unroll
    for (int i = 0; i < 64; ++i) ya[i] = 0.0f;

    const float* dxr = dx + bn * KK;
    const float* dyr = dy + bn * KK;
    const int*   ar  = adj + n * KK;
    const long   xb  = b * NN;

    #pragma unroll
    for (int k = 0; k < KK; ++k) {
        const float dxk = dxr[k];
        const float dyk = dyr[k];
        const long  xrow = (xb + (long)ar[k]) * 8;
        // this lane needs c = {2h, 2h+1, 4+2h, 5+2h}
        const float xc0 = x[xrow + 2*khalf + 0];
        const float xc1 = x[xrow + 2*khalf + 1];
        const float xc2 = x[xrow + 4 + 2*khalf + 0];
        const float xc3 = x[xrow + 4 + 2*khalf + 1];
        #pragma unroll
        for (int pd = 0; pd < 16; ++pd) {
            const float ax = dxk - mul_[pd];
            const float ay = dyk - mut_[pd];
            const float w  = norm * __expf(fmaf(ax, ax, ay*ay) * inv2s2);
            ya[(2*pd  )*2 + 0] = fmaf(w, xc0, ya[(2*pd  )*2 + 0]);
            ya[(2*pd  )*2 + 1] = fmaf(w, xc1, ya[(2*pd  )*2 + 1]);
            ya[(2*pd+1)*2 + 0] = fmaf(w, xc2, ya[(2*pd+1)*2 + 0]);
            ya[(2*pd+1)*2 + 1] = fmaf(w, xc3, ya[(2*pd+1)*2 + 1]);
        }
    }

    // ---- stage 2: 8 column tiles x 4 chained V_WMMA_F32_16X16X4_F32 ----
    #pragma unroll
    for (int j = 0; j < 8; ++j) {
        v8f acc = {};
        #pragma unroll
        for (int m = 0; m < 4; ++m) {
            v2f a;    a.x    = ya  [(j*4+m)*2+0]; a.y    = ya  [(j*4+m)*2+1];
            v2f bmat; bmat.x = ampB[(j*4+m)*2+0]; bmat.y = ampB[(j*4+m)*2+1];
            acc = __builtin_amdgcn_wmma_f32_16x16x4_f32(
                      false, a, false, bmat, (short)0, acc, false, false);
        }
        // D: VGPR v = row (v + 8*khalf), col = lane&15 of this 16-col tile
        #pragma unroll
        for (int v = 0; v < 8; ++v)
            lds[wave][v + 8*khalf][j*16 + (lane & 15)] = acc[v];
    }

    // ---- coalesced b128 stores: 16 rows x 512 B from LDS ----
    const float4* lp = (const float4*)&lds[wave][0][0];   // 512 float4s
    float4* op = (float4*)(out + bn0 * 128);
    #pragma unroll
    for (int i = 0; i < 16; ++i)
        op[i*32 + lane] = lp[i*32 + lane];
}

extern "C" void kernel_launch(void* const* d_in, const int* in_sizes, int n_in,
                              void* d_out, int out_size, void* d_ws, size_t ws_size,
                              hipStream_t stream) {
    const float* x     = (const float*)d_in[0];
    const float* dx    = (const float*)d_in[1];
    const float* dy    = (const float*)d_in[2];
    const int*   adj   = (const int*)  d_in[3];
    const float* phis  = (const float*)d_in[4];
    const float* dists = (const float*)d_in[5];
    const float* sigma = (const float*)d_in[6];
    const float* amp   = (const float*)d_in[7];
    float* out = (float*)d_out;

    const int tiles  = (BB * NN) / 16;     // 16384 waves
    const int blocks = tiles / 8;          // 8 waves per 256-thread block
    polgrid_kernel<<<blocks, 256, 0, stream>>>(x, dx, dy, adj, phis, dists,
                                               sigma, amp, out);
}